// KernelUpdateHead1D_75325136437258
// MI455X (gfx1250) — compile-verified
//
#include <hip/hip_runtime.h>
#include <hip/hip_bf16.h>
#include <math.h>

typedef __attribute__((ext_vector_type(2))) float v2f;
typedef __attribute__((ext_vector_type(4))) float v4f;
typedef __attribute__((ext_vector_type(8))) float v8f;

#define BB 16
#define NN 256
#define LL 8192
#define KK 3
#define EPSF 1e-5f

// ---------------------------------------------------------------------------
// Kernel 1: fk[b,n] = dot(initial_feats[b,n,:], features[b,:])
// 134 MB stream-once read of IF -> non-temporal loads; features stays L2-hot.
// ---------------------------------------------------------------------------
__global__ void fk_kernel(const float* __restrict__ IF, const float* __restrict__ F,
                          float* __restrict__ fk) {
    const int r   = blockIdx.x;          // b*N + n
    const int b   = r / NN;
    const int tid = threadIdx.x;         // 0..255
    const v4f* if4 = (const v4f*)(IF + (size_t)r * LL);
    const v4f* f4  = (const v4f*)(F  + (size_t)b * LL);
    float acc = 0.f;
#pragma unroll
    for (int i = 0; i < (LL / 4) / 256; ++i) {
        v4f a = __builtin_nontemporal_load(&if4[i * 256 + tid]);  // stream-once
        v4f f = f4[i * 256 + tid];                                // L2-resident
        acc += a.x * f.x + a.y * f.y + a.z * f.z + a.w * f.w;
    }
    __shared__ float s[256];
    s[tid] = acc;
    __syncthreads();
    for (int off = 128; off > 0; off >>= 1) {
        if (tid < off) s[tid] += s[tid + off];
        __syncthreads();
    }
    if (tid == 0) fk[r] = s[0];
}

// ---------------------------------------------------------------------------
// Kernel 2: tiny K=3 MLP chain + 3-head dim-1 attention. One block per batch,
// one thread per n. Negligible cost vs the two streaming stages.
// ---------------------------------------------------------------------------
struct MidParams {
    const float *fk, *kernels;
    const float *dyn_w, *dyn_b, *inp_w, *inp_b;
    const float *ig_w, *ig_b, *ug_w, *ug_b, *fc_w, *fc_b;
    const float *bn_ii_g, *bn_ii_b, *bn_ii_m, *bn_ii_v;
    const float *bn_io_g, *bn_io_b, *bn_io_m, *bn_io_v;
    const float *bn_ni_g, *bn_ni_b, *bn_ni_m, *bn_ni_v;
    const float *bn_no_g, *bn_no_b, *bn_no_m, *bn_no_v;
    const float *bn_fc_g, *bn_fc_b, *bn_fc_m, *bn_fc_v;
    const float *mha_in_w, *mha_in_b, *mha_out_w, *mha_out_b;
    const float *ln_attn_g, *ln_attn_b, *reg_w, *reg_b;
    const float *ln_reg_g, *ln_reg_b, *mask_w, *mask_b;
    float *k_new_out, *km_out;
};

__device__ __forceinline__ float bn1(float x, const float* g, const float* b,
                                     const float* m, const float* v, int c) {
    return (x - m[c]) * (g[c] * rsqrtf(v[c] + EPSF)) + b[c];
}
__device__ __forceinline__ float sigmoidf(float x) { return 1.f / (1.f + __expf(-x)); }

__global__ void mid_kernel(MidParams P) {
    const int b = blockIdx.x;
    const int n = threadIdx.x;
    const int r = b * NN + n;

    const float x = P.fk[r];
    float pi[3], po[3], kr[3], iin[3], iout[3];
#pragma unroll
    for (int j = 0; j < 3; ++j) {
        pi[j] = x * P.dyn_w[j]     + P.dyn_b[j];
        po[j] = x * P.dyn_w[3 + j] + P.dyn_b[3 + j];
        kr[j] = P.kernels[(size_t)r * 3 + j];
    }
#pragma unroll
    for (int j = 0; j < 3; ++j) {
        float s0 = P.inp_b[j], s1 = P.inp_b[3 + j];
#pragma unroll
        for (int i = 0; i < 3; ++i) {
            s0 += kr[i] * P.inp_w[j * 3 + i];
            s1 += kr[i] * P.inp_w[(3 + j) * 3 + i];
        }
        iin[j] = s0; iout[j] = s1;
    }
    float gate[3];
#pragma unroll
    for (int c = 0; c < 3; ++c) {
        float a = bn1(iin[c], P.bn_ii_g, P.bn_ii_b, P.bn_ii_m, P.bn_ii_v, c);
        float p = bn1(pi[c],  P.bn_ni_g, P.bn_ni_b, P.bn_ni_m, P.bn_ni_v, c);
        gate[c] = a * p;
    }
    float ig[3], ug[3];
#pragma unroll
    for (int j = 0; j < 3; ++j) {
        float s0 = P.ig_b[j], s1 = P.ug_b[j];
#pragma unroll
        for (int i = 0; i < 3; ++i) {
            s0 += gate[i] * P.ig_w[j * 3 + i];
            s1 += gate[i] * P.ug_w[j * 3 + i];
        }
        ig[j] = sigmoidf(bn1(s0, P.bn_ii_g, P.bn_ii_b, P.bn_ii_m, P.bn_ii_v, j));
        ug[j] = sigmoidf(bn1(s1, P.bn_ni_g, P.bn_ni_b, P.bn_ni_m, P.bn_ni_v, j));
    }
    float kb[3];
#pragma unroll
    for (int c = 0; c < 3; ++c) {
        float p = fmaxf(bn1(po[c],   P.bn_no_g, P.bn_no_b, P.bn_no_m, P.bn_no_v, c), 0.f);
        float i = fmaxf(bn1(iout[c], P.bn_io_g, P.bn_io_b, P.bn_io_m, P.bn_io_v, c), 0.f);
        kb[c] = ug[c] * p + ig[c] * i;
    }
    float kb2[3];
#pragma unroll
    for (int j = 0; j < 3; ++j) {
        float s = P.fc_b[j];
#pragma unroll
        for (int i = 0; i < 3; ++i) s += kb[i] * P.fc_w[j * 3 + i];
        kb2[j] = fmaxf(bn1(s, P.bn_fc_g, P.bn_fc_b, P.bn_fc_m, P.bn_fc_v, j), 0.f);
    }
    float qkv[9];
#pragma unroll
    for (int j = 0; j < 9; ++j) {
        float s = P.mha_in_b[j];
#pragma unroll
        for (int i = 0; i < 3; ++i) s += kb2[i] * P.mha_in_w[j * 3 + i];
        qkv[j] = s;
    }

    __shared__ float sK[3][NN];
    __shared__ float sV[3][NN];
#pragma unroll
    for (int h = 0; h < 3; ++h) { sK[h][n] = qkv[3 + h]; sV[h][n] = qkv[6 + h]; }
    __syncthreads();

    float o[3];
#pragma unroll
    for (int h = 0; h < 3; ++h) {
        const float qh = qkv[h];
        float mx = -3.402823466e38f;
        for (int m = 0; m < NN; ++m) mx = fmaxf(mx, qh * sK[h][m]);
        float se = 0.f, so = 0.f;
        for (int m = 0; m < NN; ++m) {
            float e = __expf(qh * sK[h][m] - mx);
            se += e; so += e * sV[h][m];
        }
        o[h] = so / se;
    }
    float oo[3];
#pragma unroll
    for (int j = 0; j < 3; ++j) {
        float s = P.mha_out_b[j];
#pragma unroll
        for (int i = 0; i < 3; ++i) s += o[i] * P.mha_out_w[j * 3 + i];
        oo[j] = s;
    }
    // LayerNorm(attn) -> k_new
    float knew[3];
    {
        float mu = (oo[0] + oo[1] + oo[2]) * (1.f / 3.f);
        float d0 = oo[0] - mu, d1 = oo[1] - mu, d2 = oo[2] - mu;
        float inv = rsqrtf((d0 * d0 + d1 * d1 + d2 * d2) * (1.f / 3.f) + EPSF);
        knew[0] = d0 * inv * P.ln_attn_g[0] + P.ln_attn_b[0];
        knew[1] = d1 * inv * P.ln_attn_g[1] + P.ln_attn_b[1];
        knew[2] = d2 * inv * P.ln_attn_g[2] + P.ln_attn_b[2];
    }
#pragma unroll
    for (int j = 0; j < 3; ++j) P.k_new_out[(size_t)r * 3 + j] = knew[j];

    float rl[3];
#pragma unroll
    for (int j = 0; j < 3; ++j) {
        float s = P.reg_b[j];
#pragma unroll
        for (int i = 0; i < 3; ++i) s += knew[i] * P.reg_w[j * 3 + i];
        rl[j] = s;
    }
    float rr[3];
    {
        float mu = (rl[0] + rl[1] + rl[2]) * (1.f / 3.f);
        float d0 = rl[0] - mu, d1 = rl[1] - mu, d2 = rl[2] - mu;
        float inv = rsqrtf((d0 * d0 + d1 * d1 + d2 * d2) * (1.f / 3.f) + EPSF);
        rr[0] = fmaxf(d0 * inv * P.ln_reg_g[0] + P.ln_reg_b[0], 0.f);
        rr[1] = fmaxf(d1 * inv * P.ln_reg_g[1] + P.ln_reg_b[1], 0.f);
        rr[2] = fmaxf(d2 * inv * P.ln_reg_g[2] + P.ln_reg_b[2], 0.f);
    }
#pragma unroll
    for (int j = 0; j < 3; ++j) {
        float s = P.mask_b[j];
#pragma unroll
        for (int i = 0; i < 3; ++i) s += rr[i] * P.mask_w[j * 3 + i];
        P.km_out[(size_t)r * 3 + j] = s;
    }
}

// ---------------------------------------------------------------------------
// Kernel 3: new_feats[b,n,l] = sum_t km[b,n,t] * features[b, l+t-1]
// V_WMMA_F32_16X16X4_F32: C(16 l x 16 n) = A(16x4 features taps) x B(4x16 km).
// Each wave handles 4 consecutive 16-l tiles (64 l, 4 KB of output):
//  - B (km taps) loaded once per wave, reused by all 4 WMMAs
//  - A boundary handled branch-free (clamp + select, no exec-mask branches)
//  - output written with non-temporal float4 stores (stream-once, skip L2 fill)
// ---------------------------------------------------------------------------
__global__ void conv_wmma_kernel(const float* __restrict__ F, const float* __restrict__ km,
                                 float* __restrict__ out) {
    const int wave   = threadIdx.x >> 5;
    const int lane   = threadIdx.x & 31;
    const int l_base = (blockIdx.x * 8 + wave) * 64;   // 4 tiles of 16 l
    const int n0     = blockIdx.y * 16;
    const int b      = blockIdx.z;

    const float* f   = F  + (size_t)b * LL;
    const float* kmb = km + ((size_t)b * NN + n0) * KK;

    const int  M  = lane & 15;
    const bool hi = lane >= 16;

    // B operand: K taps x 16 n-cols of km; invariant across the 4 l-tiles.
    // B layout: vgpr0 = rows K0 (lanes 0-15) / K2 (lanes 16-31), vgpr1 = K1 / K3(pad).
    v2f bm;
    bm.x = hi ? kmb[M * 3 + 2] : kmb[M * 3 + 0];
    bm.y = hi ? 0.f            : kmb[M * 3 + 1];

    float* dstbase = out + ((size_t)b * NN + n0 + M) * LL;

#pragma unroll
    for (int t = 0; t < 4; ++t) {
        const int l0 = l_base + t * 16;
        // A: 16 l-rows x K taps (t=0 -> l-1, t=1 -> l, t=2 -> l+1, t=3 -> pad)
        // lanes 0-15 hold K0,K1 in vgpr0,1; lanes 16-31 hold K2,K3.
        const int la = l0 + M + (hi ? 1 : -1);
        const int lc = min(max(la, 0), LL - 1);        // branch-free edge handling
        const float fa = f[lc];
        v2f a;
        a.x = (la == lc) ? fa : 0.f;                   // K=0 (lo) / K=2 (hi)
        a.y = hi ? 0.f : f[l0 + M];                    // K=1 (lo) / K=3 pad (hi)

        v8f c = {};
        c = __builtin_amdgcn_wmma_f32_16x16x4_f32(false, a, false, bm,
                                                  (short)0, c, false, false);

        // C layout: vgpr v, lanes 0-15 -> row v; lanes 16-31 -> row v+8;
        // column n = n0 + (lane&15). Rows are consecutive l -> two float4s.
        float* dst = dstbase + l0 + (hi ? 8 : 0);
        v4f v0; v0.x = c[0]; v0.y = c[1]; v0.z = c[2]; v0.w = c[3];
        v4f v1; v1.x = c[4]; v1.y = c[5]; v1.z = c[6]; v1.w = c[7];
        __builtin_nontemporal_store(v0, (v4f*)dst);
        __builtin_nontemporal_store(v1, (v4f*)dst + 1);
    }
}

// ---------------------------------------------------------------------------
extern "C" void kernel_launch(void* const* d_in, const int* in_sizes, int n_in,
                              void* d_out, int out_size, void* d_ws, size_t ws_size,
                              hipStream_t stream) {
    const float* features = (const float*)d_in[0];
    const float* kernels  = (const float*)d_in[1];
    const float* initial  = (const float*)d_in[2];
    float* out = (float*)d_out;

    float* fk = (float*)d_ws;          // B*N floats
    float* km = fk + BB * NN;          // B*N*K floats

    fk_kernel<<<BB * NN, 256, 0, stream>>>(initial, features, fk);

    MidParams P;
    P.fk = fk; P.kernels = kernels;
    P.dyn_w = (const float*)d_in[3];  P.dyn_b = (const float*)d_in[4];
    P.inp_w = (const float*)d_in[5];  P.inp_b = (const float*)d_in[6];
    P.ig_w  = (const float*)d_in[7];  P.ig_b  = (const float*)d_in[8];
    P.ug_w  = (const float*)d_in[9];  P.ug_b  = (const float*)d_in[10];
    P.fc_w  = (const float*)d_in[11]; P.fc_b  = (const float*)d_in[12];
    P.bn_ii_g = (const float*)d_in[13]; P.bn_ii_b = (const float*)d_in[14];
    P.bn_ii_m = (const float*)d_in[15]; P.bn_ii_v = (const float*)d_in[16];
    P.bn_io_g = (const float*)d_in[17]; P.bn_io_b = (const float*)d_in[18];
    P.bn_io_m = (const float*)d_in[19]; P.bn_io_v = (const float*)d_in[20];
    P.bn_ni_g = (const float*)d_in[21]; P.bn_ni_b = (const float*)d_in[22];
    P.bn_ni_m = (const float*)d_in[23]; P.bn_ni_v = (const float*)d_in[24];
    P.bn_no_g = (const float*)d_in[25]; P.bn_no_b = (const float*)d_in[26];
    P.bn_no_m = (const float*)d_in[27]; P.bn_no_v = (const float*)d_in[28];
    P.bn_fc_g = (const float*)d_in[29]; P.bn_fc_b = (const float*)d_in[30];
    P.bn_fc_m = (const float*)d_in[31]; P.bn_fc_v = (const float*)d_in[32];
    P.mha_in_w  = (const float*)d_in[33]; P.mha_in_b  = (const float*)d_in[34];
    P.mha_out_w = (const float*)d_in[35]; P.mha_out_b = (const float*)d_in[36];
    P.ln_attn_g = (const float*)d_in[37]; P.ln_attn_b = (const float*)d_in[38];
    P.reg_w = (const float*)d_in[39]; P.reg_b = (const float*)d_in[40];
    P.ln_reg_g = (const float*)d_in[41]; P.ln_reg_b = (const float*)d_in[42];
    P.mask_w = (const float*)d_in[43]; P.mask_b = (const float*)d_in[44];
    P.k_new_out = out + (size_t)BB * NN * LL;   // k_new follows new_feats
    P.km_out = km;

    mid_kernel<<<BB, NN, 0, stream>>>(P);

    // grid: (L/512, N/16, B) ; 8 waves/block, each wave = 64 l x 16 n tile strip
    conv_wmma_kernel<<<dim3(LL / 512, NN / 16, BB), 256, 0, stream>>>(features, km, out);
}